// MatrixContrastiveLoss_55499567399443
// MI455X (gfx1250) — compile-verified
//
#include <hip/hip_runtime.h>
#include <hip/hip_bf16.h>
#include <math.h>

// Problem constants from the reference
#define NB   8          // N batches
#define NG   16         // NM groups per batch
#define GM   16         // M samples per group
#define EE   64         // embedding dim
#define S    (NG * GM)  // 256 samples per batch
#define TILES_PER_BATCH ((S / 16) * (S / 16))   // 16x16 tiles of the SxS matrix = 256
#define TOTAL_TILES (NB * TILES_PER_BATCH)      // 2048

typedef float v2f __attribute__((ext_vector_type(2)));
typedef float v8f __attribute__((ext_vector_type(8)));

// ---------------------------------------------------------------------------
// Kernel 1: per-sample squared norms  s_i = ||x_i||^2   (2048 values)
//           + zero the 16 per-batch accumulators (totals[8] + matches[8])
// ---------------------------------------------------------------------------
__global__ __launch_bounds__(256) void mcl_norms_init(
    const float* __restrict__ x, float* __restrict__ s, float* __restrict__ acc) {
  int i = blockIdx.x * blockDim.x + threadIdx.x;   // 0 .. NB*S-1
  if (i < 16) acc[i] = 0.0f;
  if (i >= NB * S) return;
  const float* xp = x + (size_t)i * EE;
  float sum = 0.0f;
#pragma unroll
  for (int k = 0; k < EE; k += 4) {
    float4 v = *(const float4*)(xp + k);
    sum = fmaf(v.x, v.x, sum);
    sum = fmaf(v.y, v.y, sum);
    sum = fmaf(v.z, v.z, sum);
    sum = fmaf(v.w, v.w, sum);
  }
  s[i] = sum;
}

// ---------------------------------------------------------------------------
// Kernel 2: one wave per 16x16 Gram tile.
//   G = X_tile_i · X_tile_j^T via 16 chained V_WMMA_F32_16X16X4_F32 (K=4 each).
//   ISA layouts (CDNA5 ISA):
//     A 16x4 f32 : lane l<16 -> (M=l,   K=k0..k0+1), lane l>=16 -> (M=l-16, K=k0+2..k0+3)
//     B 4x16 f32 : lane n<16 -> (N=n,   K=k0..k0+1), lane n>=16 -> (N=n-16, K=k0+2..k0+3)
//   => for the Gram product both operands load identically: 8B per lane per step.
//   C/D layout: VGPR v, lane<16 -> (M=v, N=lane); lane>=16 -> (M=v+8, N=lane-16).
// ---------------------------------------------------------------------------
__global__ __launch_bounds__(256) void mcl_gram_tiles(
    const float* __restrict__ x, const float* __restrict__ s,
    float* __restrict__ acc /* [0..7]=totals, [8..15]=matches */) {
  const int wave = (blockIdx.x * blockDim.x + threadIdx.x) >> 5;  // global tile id
  const int lane = threadIdx.x & 31;
  if (wave >= TOTAL_TILES) return;   // grid sized exactly; keeps EXEC full in practice

  const int n  = wave >> 8;          // batch
  const int ti = (wave >> 4) & 15;   // row tile
  const int tj = wave & 15;          // col tile

  const float* xb = x + (size_t)n * S * EE;
  const float* sb = s + n * S;

  const int half = lane >> 4;        // 0: K+{0,1}, 1: K+{2,3}
  const int lr   = lane & 15;
  const float* ap = xb + (size_t)(ti * 16 + lr) * EE + 2 * half;  // A rows
  const float* bp = xb + (size_t)(tj * 16 + lr) * EE + 2 * half;  // B cols (= rows of X)

  v8f c = {};
#pragma unroll
  for (int k0 = 0; k0 < EE; k0 += 4) {
    v2f a = *(const v2f*)(ap + k0);
    v2f b = *(const v2f*)(bp + k0);
    // 8 args: (neg_a, A, neg_b, B, c_mod, C, reuse_a, reuse_b)
    c = __builtin_amdgcn_wmma_f32_16x16x4_f32(false, a, false, b, (short)0, c,
                                              false, false);
  }

  // Squared norms for this lane's rows/col
  const float scol = sb[tj * 16 + lr];
  const float* sr  = sb + ti * 16 + 8 * half;
  const int gcol = tj * 16 + lr;
  const int growbase = ti * 16 + 8 * half;

  float partial = 0.0f;
#pragma unroll
  for (int v = 0; v < 8; ++v) {
    float d2 = fmaxf(sr[v] + scol - 2.0f * c[v], 0.0f);
    float kv = ((growbase + v) == gcol) ? 1.0f : expf(-0.5f * sqrtf(d2));
    partial += kv;
  }

  // wave32 reduction
#pragma unroll
  for (int off = 16; off >= 1; off >>= 1)
    partial += __shfl_xor(partial, off, 32);

  if (lane == 0) {
    atomicAdd(&acc[n], partial);                 // total kernel sum
    if (ti == tj) atomicAdd(&acc[8 + n], partial);  // matches = diagonal 16x16 blocks
  }
}

// ---------------------------------------------------------------------------
// Kernel 3: combine the 16 per-batch scalars into the final loss.
// ---------------------------------------------------------------------------
__global__ void mcl_finalize(const float* __restrict__ acc, float* __restrict__ out) {
  if (threadIdx.x != 0 || blockIdx.x != 0) return;
  const float inv_m  = 1.0f / (float)(NG * GM * GM);           // 1/4096
  const float inv_nm = 1.0f / (float)(S * (S - GM));           // 1/61440
  float csum = 0.0f, ms = 0.0f, nms = 0.0f;
  for (int n = 0; n < NB; ++n) {
    float m  = acc[n + 8] * inv_m;
    float nm = (acc[n] - acc[n + 8]) * inv_nm;
    csum += fmaxf(nm - m + 0.5f, 0.0f);   // MARGIN = 0.5
    ms += m;
    nms += nm;
  }
  const float invN = 1.0f / (float)NB;
  // W_CONTRASTIVE * contrastive + W_MATCHES * (1 - matches) + W_NON_MATCHES * nonmatches
  out[0] = csum * invN + (1.0f - ms * invN) + nms * invN;
}

// ---------------------------------------------------------------------------
extern "C" void kernel_launch(void* const* d_in, const int* in_sizes, int n_in,
                              void* d_out, int out_size, void* d_ws, size_t ws_size,
                              hipStream_t stream) {
  const float* x = (const float*)d_in[0];     // [8,16,16,64] fp32
  float* out = (float*)d_out;                  // scalar fp32

  float* ws   = (float*)d_ws;
  float* s    = ws;                // 2048 floats: per-sample squared norms
  float* acc  = ws + NB * S;       // 16 floats: totals[8], matches[8]

  // 1) norms + zero accumulators (NB*S = 2048 threads)
  mcl_norms_init<<<(NB * S + 255) / 256, 256, 0, stream>>>(x, s, acc);

  // 2) WMMA Gram tiles: 2048 waves, 8 waves (256 threads) per block
  mcl_gram_tiles<<<TOTAL_TILES / 8, 256, 0, stream>>>(x, s, acc);

  // 3) finalize scalar
  mcl_finalize<<<1, 32, 0, stream>>>(acc, out);
}